// ContrastiveLoss_33870112096257
// MI455X (gfx1250) — compile-verified
//
#include <hip/hip_runtime.h>

#define Bsz   4
#define Edim  16
#define Npix  (512 * 512)
#define Cn    24
#define DELTA_VAR  0.75f
#define DELTA_DIST 2.0f
#define ALPHA 1.0f
#define BETA  1.0f
#define GAMMA 0.001f

typedef float v2f __attribute__((ext_vector_type(2)));
typedef float v8f __attribute__((ext_vector_type(8)));

// ---------------- ws layout (floats) ----------------
// sums  : [B][C][16]  -> 1536 @ 0
// counts: [B][C]      ->   96 @ 1536
// mu    : [B][C][16]  -> 1536 @ 1632
// hsum  : [B][C]      ->   96 @ 3168
// total 3264 floats

__global__ void dl_zero_ws(float* ws, int n) {
    int i = blockIdx.x * blockDim.x + threadIdx.x;
    if (i < n) ws[i] = 0.f;
}

// Pass 1: per-(b,c) embedding sums and counts. 1024 blocks, 256 threads,
// 4 pixels/thread via float4 loads (fully coalesced 512B/wave/load).
__global__ void dl_accum_sums(const float* __restrict__ inp, const int* __restrict__ tgt,
                              float* __restrict__ sums, float* __restrict__ counts) {
    __shared__ float accS[Cn * 17];   // padded stride 17 -> conflict-free banks
    __shared__ float accN[Cn];
    const int t = threadIdx.x;
    for (int i = t; i < Cn * 17; i += 256) accS[i] = 0.f;
    if (t < Cn) accN[t] = 0.f;
    __syncthreads();

    const int b     = blockIdx.x >> 8;
    const int chunk = blockIdx.x & 255;
    const int n0    = chunk * 1024 + t * 4;

    const int4  lab   = *reinterpret_cast<const int4*>(tgt + (size_t)b * Npix + n0);
    const float* base = inp + (size_t)b * Edim * Npix + n0;

#pragma unroll
    for (int e = 0; e < Edim; e++) {
        float4 v = *reinterpret_cast<const float4*>(base + (size_t)e * Npix);
        atomicAdd(&accS[lab.x * 17 + e], v.x);
        atomicAdd(&accS[lab.y * 17 + e], v.y);
        atomicAdd(&accS[lab.z * 17 + e], v.z);
        atomicAdd(&accS[lab.w * 17 + e], v.w);
    }
    atomicAdd(&accN[lab.x], 1.f);
    atomicAdd(&accN[lab.y], 1.f);
    atomicAdd(&accN[lab.z], 1.f);
    atomicAdd(&accN[lab.w], 1.f);
    __syncthreads();

    for (int i = t; i < Cn * 17; i += 256) {
        int c = i / 17, e = i % 17;
        if (e < 16) atomicAdd(&sums[(b * Cn + c) * 16 + e], accS[i]);
    }
    if (t < Cn) atomicAdd(&counts[b * Cn + t], accN[t]);
}

__global__ void dl_compute_mu(const float* __restrict__ sums,
                              const float* __restrict__ counts, float* __restrict__ mu) {
    int i = blockIdx.x * 256 + threadIdx.x;
    if (i < Bsz * Cn * 16) mu[i] = sums[i] / counts[i >> 4];
}

// Pass 2: hinged distance of each pixel to its own cluster mean, segment-summed.
__global__ void dl_accum_var(const float* __restrict__ inp, const int* __restrict__ tgt,
                             const float* __restrict__ mu, float* __restrict__ hsum) {
    __shared__ float mu_s[Cn * 17];
    __shared__ float hacc[Cn];
    const int t = threadIdx.x;
    const int b = blockIdx.x >> 8;
    for (int i = t; i < Cn * 17; i += 256) {
        int c = i / 17, e = i % 17;
        mu_s[i] = (e < 16) ? mu[(b * Cn + c) * 16 + e] : 0.f;
    }
    if (t < Cn) hacc[t] = 0.f;
    __syncthreads();

    const int chunk = blockIdx.x & 255;
    const int n0    = chunk * 1024 + t * 4;
    const int4  lab   = *reinterpret_cast<const int4*>(tgt + (size_t)b * Npix + n0);
    const float* base = inp + (size_t)b * Edim * Npix + n0;

    float d2x = 0.f, d2y = 0.f, d2z = 0.f, d2w = 0.f;
#pragma unroll
    for (int e = 0; e < Edim; e++) {
        float4 v = *reinterpret_cast<const float4*>(base + (size_t)e * Npix);
        float dx = v.x - mu_s[lab.x * 17 + e]; d2x += dx * dx;
        float dy = v.y - mu_s[lab.y * 17 + e]; d2y += dy * dy;
        float dz = v.z - mu_s[lab.z * 17 + e]; d2z += dz * dz;
        float dw = v.w - mu_s[lab.w * 17 + e]; d2w += dw * dw;
    }
    {
        float h;
        h = fmaxf(sqrtf(d2x) - DELTA_VAR, 0.f); atomicAdd(&hacc[lab.x], h * h);
        h = fmaxf(sqrtf(d2y) - DELTA_VAR, 0.f); atomicAdd(&hacc[lab.y], h * h);
        h = fmaxf(sqrtf(d2z) - DELTA_VAR, 0.f); atomicAdd(&hacc[lab.z], h * h);
        h = fmaxf(sqrtf(d2w) - DELTA_VAR, 0.f); atomicAdd(&hacc[lab.w], h * h);
    }
    __syncthreads();
    if (t < Cn) atomicAdd(&hsum[b * Cn + t], hacc[t]);
}

// Single-wave finalize. mu[b] is staged into LDS zero-padded to 32x16 so every
// WMMA fragment is a uniform, unconditional ds_load_b64 (no exec-mask
// divergence around the WMMAs).
//
// V_WMMA_F32_16X16X4_F32 A 16x4 layout: lanes 0-15 hold M=lane, K={k0,k0+1};
// lanes 16-31 hold M=lane-16, K={k0+2,k0+3}. For G = M*M^T the B fragment
// (4x16 = M^T chunk) is lane-identical to the A fragment of the column tile.
__global__ void dl_finalize(const float* __restrict__ mu, const float* __restrict__ counts,
                            const float* __restrict__ hsum, float* __restrict__ out) {
    __shared__ float mu_pad[32 * 16];   // rows 24..31 zero
    __shared__ float G[32 * 33];
    __shared__ float red[3];
    const int lane = threadIdx.x;

    float var_acc = 0.f, dist_acc = 0.f, reg_acc = 0.f;
    for (int i = lane; i < Bsz * Cn; i += 32) var_acc += hsum[i] / counts[i];

    for (int b = 0; b < Bsz; b++) {
        const float* mu_b = mu + b * Cn * 16;
        for (int i = lane; i < 32 * 16; i += 32)
            mu_pad[i] = (i < Cn * 16) ? mu_b[i] : 0.f;
        __syncthreads();

        const int r  = lane & 15;
        const int kh = (lane >> 4) << 1;   // 0 or 2

        v8f acc[4];
#pragma unroll
        for (int rt = 0; rt < 2; rt++) {
#pragma unroll
            for (int ct = 0; ct < 2; ct++) {
                v8f a = {};
#pragma unroll
                for (int k0 = 0; k0 < 16; k0 += 4) {
                    v2f av = *reinterpret_cast<const v2f*>(&mu_pad[(rt * 16 + r) * 16 + k0 + kh]);
                    v2f bv = *reinterpret_cast<const v2f*>(&mu_pad[(ct * 16 + r) * 16 + k0 + kh]);
                    a = __builtin_amdgcn_wmma_f32_16x16x4_f32(
                            false, av, false, bv, (short)0, a, false, false);
                }
                acc[rt * 2 + ct] = a;
            }
        }
        // scatter D fragments: vgpr v holds row (lane<16 ? v : 8+v), col lane&15
#pragma unroll
        for (int rt = 0; rt < 2; rt++) {
#pragma unroll
            for (int ct = 0; ct < 2; ct++) {
                v8f a = acc[rt * 2 + ct];
                int col   = ct * 16 + r;
                int rbase = rt * 16 + ((lane >> 4) ? 8 : 0);
#pragma unroll
                for (int v = 0; v < 8; v++) G[(rbase + v) * 33 + col] = a[v];
            }
        }
        __syncthreads();
        for (int p = lane; p < Cn * Cn; p += 32) {
            int i = p / Cn, j = p % Cn;
            float gii = G[i * 33 + i], gjj = G[j * 33 + j], gij = G[i * 33 + j];
            float dd = fmaxf(gii + gjj - 2.f * gij, 0.f);
            if (i != j) {
                float h = fmaxf(2.f * DELTA_DIST - sqrtf(dd), 0.f);
                dist_acc += h * h;
            }
        }
        for (int i = lane; i < Cn; i += 32) reg_acc += sqrtf(fmaxf(G[i * 33 + i], 0.f));
        __syncthreads();
    }

    if (lane == 0) { red[0] = 0.f; red[1] = 0.f; red[2] = 0.f; }
    __syncthreads();
    atomicAdd(&red[0], var_acc);
    atomicAdd(&red[1], dist_acc);
    atomicAdd(&red[2], reg_acc);
    __syncthreads();
    if (lane == 0) {
        float var = red[0] / (float)(Bsz * Cn);
        float dis = red[1] / (float)(Bsz * Cn * (Cn - 1));
        float reg = red[2] / (float)(Bsz * Cn);
        out[0] = ALPHA * var + BETA * dis + GAMMA * reg;
    }
}

extern "C" void kernel_launch(void* const* d_in, const int* in_sizes, int n_in,
                              void* d_out, int out_size, void* d_ws, size_t ws_size,
                              hipStream_t stream) {
    (void)in_sizes; (void)n_in; (void)out_size; (void)ws_size;
    const float* inp = (const float*)d_in[0];
    const int*   tgt = (const int*)d_in[1];
    float* out = (float*)d_out;
    float* ws  = (float*)d_ws;

    float* sums   = ws;            // 1536
    float* counts = ws + 1536;     //   96
    float* mu     = ws + 1632;     // 1536
    float* hsum   = ws + 3168;     //   96

    dl_zero_ws<<<13, 256, 0, stream>>>(ws, 3264);
    dl_accum_sums<<<Bsz * 256, 256, 0, stream>>>(inp, tgt, sums, counts);
    dl_compute_mu<<<6, 256, 0, stream>>>(sums, counts, mu);
    dl_accum_var<<<Bsz * 256, 256, 0, stream>>>(inp, tgt, mu, hsum);
    dl_finalize<<<1, 32, 0, stream>>>(mu, counts, hsum, out);
}